// ChildSumTreeLSTM_29154238005592
// MI455X (gfx1250) — compile-verified
//
#include <hip/hip_runtime.h>

// ---------------- CDNA5 (gfx1250, wave32) Child-Sum Tree-LSTM ----------------
// fp32 WMMA path: V_WMMA_F32_16X16X4_F32, D(16x16,f32) = A(16x4) x B(4x16) + C.
// A operand (2 VGPRs): lane l holds row M=l&15; VGPR0 = K = ((l>=16)?2:0),
//                      VGPR1 = K+1.
// B operand (2 VGPRs): lane l holds col N=l&15; VGPR0 = K = ((l>=16)?2:0),
//                      VGPR1 = K+1.   (mirrors documented 32-bit A layout)
// C/D (8 VGPRs):       VGPR v = row v (lanes 0-15) / row v+8 (lanes 16-31).

#define ROWS       16
#define SA_STRIDE  132   // LDS row stride (floats): pad 4 -> conflict-free b64 frag loads

typedef float v2f __attribute__((ext_vector_type(2)));
typedef float v8f __attribute__((ext_vector_type(8)));

__device__ __forceinline__ float sigm(float x) { return 1.0f / (1.0f + expf(-x)); }

// K=128 GEMM micro-tile: 32 chained WMMA f32 16x16x4 steps.
// aBase = &LDS_A[(m)*SA_STRIDE + kh]  (per-lane), bTile = swizzled weight tile.
__device__ __forceinline__ v8f wmma_k128(const float* __restrict__ aBase,
                                         const v2f*   __restrict__ bTile,
                                         int lane)
{
    v8f acc = (v8f)(0.0f);
#pragma unroll
    for (int ks = 0; ks < 32; ++ks) {
        v2f a = *(const v2f*)(aBase + ks * 4);
        v2f b = bTile[ks * 32 + lane];
        acc = __builtin_amdgcn_wmma_f32_16x16x4_f32(
                  false, a, false, b, (short)0, acc, false, false);
    }
    return acc;
}

// Pre-swizzle a KxN (K=128) fp32 weight matrix into per-lane fragment order:
// out float2[nt][ks][lane] = { W[k][n], W[k+1][n] },
//   k = ks*4 + 2*(lane>=16), n = nt*16 + (lane&15)
__global__ void swizzle_kernel(const float* __restrict__ W, float* __restrict__ out,
                               int N, int ntiles)
{
    int total = ntiles * 1024;                      // 32 ksteps * 32 lanes per tile
    for (int idx = blockIdx.x * blockDim.x + threadIdx.x; idx < total;
         idx += gridDim.x * blockDim.x) {
        int nt  = idx >> 10;
        int rem = idx & 1023;
        int ks  = rem >> 5;
        int l   = rem & 31;
        int k   = ks * 4 + ((l >> 4) << 1);
        int nn  = nt * 16 + (l & 15);
        out[idx * 2 + 0] = W[k * N + nn];
        out[idx * 2 + 1] = W[(k + 1) * N + nn];
    }
}

// B_ext[m][d] = battnh[d] + sum_k h_ext[m][k] * Wattnh[128+k][d]   (16x128)
__global__ void bext_kernel(const float* __restrict__ hext,
                            const float* __restrict__ Wattnh,
                            const float* __restrict__ battnh,
                            float* __restrict__ bext)
{
    int mrow = blockIdx.x;      // 0..15
    int d    = threadIdx.x;     // 0..127
    float acc = battnh[d];
    for (int k = 0; k < 128; ++k)
        acc += hext[mrow * 128 + k] * Wattnh[(128 + k) * 128 + d];
    bext[mrow * 128 + d] = acc;
}

// One tree level (or the leaf level). 256 threads = 8 wave32 per 16-row tile.
__global__ void __launch_bounds__(256)
tree_level_kernel(const float* __restrict__ x,       // leaf input rows (leaf only)
                  const float* __restrict__ hIn,     // children h (internal only)
                  const float* __restrict__ cIn,     // children c (internal only)
                  float* __restrict__ cOut,          // this level's c (ws)
                  float* __restrict__ outBase,       // d_out base
                  int hOutOff,                       // row offset of this level's h
                  int n,                             // nodes produced this level
                  const float* __restrict__ wsWiou,  // swizzled Wioux / Wiouh
                  const float* __restrict__ wsWfh,   // swizzled Wfh (internal only)
                  const float* __restrict__ wsWat,   // swizzled Wattnh[:128]
                  const float* __restrict__ bias1,   // bioux (leaf) or null
                  const float* __restrict__ bias2,   // biouh
                  const float* __restrict__ bfh,
                  const float* __restrict__ bext,    // precomputed B_ext (ws)
                  const float* __restrict__ hext,    // h_ext
                  const float* __restrict__ wa,
                  int isLeaf, int isFinal)
{
    const int tid  = threadIdx.x;
    const int lane = tid & 31;
    const int w    = tid >> 5;               // wave id 0..7
    const int row0 = blockIdx.x * ROWS;

    __shared__ float sA  [32 * SA_STRIDE];   // children h (internal)
    __shared__ float sX  [ROWS * SA_STRIDE]; // x rows (leaf) or child-sum h
    __shared__ float sIOU[ROWS * 384];
    __shared__ float sF  [32 * 128];
    __shared__ float sH  [ROWS * SA_STRIDE]; // pre-attention h
    __shared__ float sHW [ROWS * 128];       // h @ Wattnh_top
    __shared__ float sBext[16 * 128];
    __shared__ float sHext[16 * 128];
    __shared__ float sWa [128];
    __shared__ float sSc [16 * 16];          // logits, then softmax weights
    __shared__ float sSsum[16];

    // ---- stage ----
    for (int i = tid; i < 16 * 128; i += 256) { sBext[i] = bext[i]; sHext[i] = hext[i]; }
    if (tid < 128) sWa[tid] = wa[tid];

    if (isLeaf) {
        for (int i = tid; i < ROWS * 128; i += 256) {
            int r = i >> 7, d = i & 127;
            sX[r * SA_STRIDE + d] = x[(size_t)(row0 + r) * 128 + d];
        }
    } else {
        int crows = 2 * n;
        for (int i = tid; i < 32 * 128; i += 256) {
            int r = i >> 7, d = i & 127;
            int cr = 2 * row0 + r; if (cr > crows - 1) cr = crows - 1;
            sA[r * SA_STRIDE + d] = hIn[(size_t)cr * 128 + d];
        }
    }
    __syncthreads();

    if (!isLeaf) {  // child-sum
        for (int i = tid; i < ROWS * 128; i += 256) {
            int r = i >> 7, d = i & 127;
            sX[r * SA_STRIDE + d] = sA[(2 * r) * SA_STRIDE + d]
                                  + sA[(2 * r + 1) * SA_STRIDE + d];
        }
        __syncthreads();
    }

    // ---- GEMMs (wave-uniform control flow; EXEC stays all-ones at WMMA) ----
    const int m  = lane & 15;
    const int kh = (lane >> 4) << 1;
    const int rb = (lane >> 4) << 3;
    const float* aX = sX + m * SA_STRIDE + kh;

    // iou: [16,128] @ [128,384]  -> 24 N-tiles, 3 per wave
#pragma unroll
    for (int t = 0; t < 3; ++t) {
        int nt = w + t * 8;
        v8f acc = wmma_k128(aX, (const v2f*)wsWiou + (size_t)nt * 1024, lane);
        int col = nt * 16 + m;
        float b = bias2[col] + (bias1 ? bias1[col] : 0.0f);
#pragma unroll
        for (int v = 0; v < 8; ++v) sIOU[(rb + v) * 384 + col] = acc[v] + b;
    }

    if (!isLeaf) {  // f: [32,128] @ [128,128] -> 2 M-tiles x 8 N-tiles, 2 per wave
#pragma unroll
        for (int mt = 0; mt < 2; ++mt) {
            const float* aF = sA + (mt * 16 + m) * SA_STRIDE + kh;
            v8f acc = wmma_k128(aF, (const v2f*)wsWfh + (size_t)w * 1024, lane);
            int col = w * 16 + m;
            float b = bfh[col];
#pragma unroll
            for (int v = 0; v < 8; ++v)
                sF[(mt * 16 + rb + v) * 128 + col] = acc[v] + b;
        }
    }
    __syncthreads();

    // ---- gates ----
    for (int i = tid; i < ROWS * 128; i += 256) {
        int r = i >> 7, d = i & 127;
        float iv = sIOU[r * 384 + d];
        float ov = sIOU[r * 384 + 128 + d];
        float uv = sIOU[r * 384 + 256 + d];
        float cn = sigm(iv) * tanhf(uv);
        if (!isLeaf) {
            int crows = 2 * n;
            int c0 = 2 * (row0 + r), c1 = c0 + 1;
            if (c0 > crows - 1) c0 = crows - 1;
            if (c1 > crows - 1) c1 = crows - 1;
            cn += sigm(sF[(2 * r) * 128 + d])     * cIn[(size_t)c0 * 128 + d]
                + sigm(sF[(2 * r + 1) * 128 + d]) * cIn[(size_t)c1 * 128 + d];
        }
        float hv = sigm(ov) * tanhf(cn);
        if (row0 + r < n) {
            cOut[(size_t)(row0 + r) * 128 + d] = cn;
            if (isFinal) outBase[(size_t)(row0 + r) * 128 + d] = cn;  // final c -> rows 0..1
        }
        sH[r * SA_STRIDE + d] = hv;
    }
    __syncthreads();

    // ---- attention: HW = h @ Wattnh_top, 8 N-tiles, 1 per wave ----
    {
        const float* aH = sH + m * SA_STRIDE + kh;
        v8f acc = wmma_k128(aH, (const v2f*)wsWat + (size_t)w * 1024, lane);
        int col = w * 16 + m;
#pragma unroll
        for (int v = 0; v < 8; ++v) sHW[(rb + v) * 128 + col] = acc[v];
    }
    __syncthreads();

    // logits: thread (r,mm) -> sum_d tanh(HW[r][d] + Bext[mm][d]) * Wa[d]
    {
        int r = tid >> 4, mm = tid & 15;
        const float* hwRow = sHW + r * 128;
        const float* beRow = sBext + mm * 128;
        float acc = 0.0f;
#pragma unroll 4
        for (int d = 0; d < 128; ++d)
            acc += tanhf(hwRow[d] + beRow[d]) * sWa[d];
        sSc[r * 16 + mm] = acc;
    }
    __syncthreads();

    if (tid < 16) {  // per-row softmax over m=16
        float mx = -1e30f;
        for (int j = 0; j < 16; ++j) mx = fmaxf(mx, sSc[tid * 16 + j]);
        float e[16], sum = 0.0f;
        for (int j = 0; j < 16; ++j) { e[j] = expf(sSc[tid * 16 + j] - mx); sum += e[j]; }
        float inv = 1.0f / sum, ss = 0.0f;
        for (int j = 0; j < 16; ++j) { float sv = e[j] * inv; sSc[tid * 16 + j] = sv; ss += sv; }
        sSsum[tid] = ss;
    }
    __syncthreads();

    // out_h = ssum*h + sum_m (1-s_m) * h_ext_m  -> d_out (doubles as next level input)
    for (int i = tid; i < ROWS * 128; i += 256) {
        int r = i >> 7, d = i & 127;
        float acc = sSsum[r] * sH[r * SA_STRIDE + d];
#pragma unroll
        for (int mm = 0; mm < 16; ++mm)
            acc += (1.0f - sSc[r * 16 + mm]) * sHext[mm * 128 + d];
        if (row0 + r < n) {
            outBase[(size_t)(hOutOff + row0 + r) * 128 + d] = acc;
            if (isFinal) outBase[(size_t)(2 + row0 + r) * 128 + d] = acc;  // final h -> rows 2..3
        }
    }
}

extern "C" void kernel_launch(void* const* d_in, const int* in_sizes, int n_in,
                              void* d_out, int out_size, void* d_ws, size_t ws_size,
                              hipStream_t stream)
{
    const float* x      = (const float*)d_in[0];
    const float* h_ext  = (const float*)d_in[1];
    const float* Wioux  = (const float*)d_in[2];
    const float* bioux  = (const float*)d_in[3];
    const float* Wiouh  = (const float*)d_in[4];
    const float* biouh  = (const float*)d_in[5];
    const float* Wfh    = (const float*)d_in[6];
    const float* bfh    = (const float*)d_in[7];
    const float* Wattnh = (const float*)d_in[8];
    const float* battnh = (const float*)d_in[9];
    const float* Wa     = (const float*)d_in[10];
    float* out = (float*)d_out;
    float* ws  = (float*)d_ws;

    // ws layout (floats)
    float* wsWioux = ws + 0;                     // 24*1024*2 = 49152
    float* wsWiouh = ws + 49152;                 // 49152
    float* wsWfh   = ws + 98304;                 // 8*1024*2 = 16384
    float* wsWat   = ws + 114688;                // 16384
    float* wsBext  = ws + 131072;                // 2048
    float* cA      = ws + 133120;                // 16384*128
    float* cB      = ws + 133120 + 16384 * 128;  // 8192*128

    // one-time (per launch) weight swizzle + attention-bias precompute
    swizzle_kernel<<<96, 256, 0, stream>>>(Wioux,  wsWioux, 384, 24);
    swizzle_kernel<<<96, 256, 0, stream>>>(Wiouh,  wsWiouh, 384, 24);
    swizzle_kernel<<<32, 256, 0, stream>>>(Wfh,    wsWfh,   128, 8);
    swizzle_kernel<<<32, 256, 0, stream>>>(Wattnh, wsWat,   128, 8);  // top 128 rows
    bext_kernel<<<16, 128, 0, stream>>>(h_ext, Wattnh, battnh, wsBext);

    // leaf level: n = 16384, h rows at offset 4
    int off = 4;
    tree_level_kernel<<<16384 / 16, 256, 0, stream>>>(
        x, nullptr, nullptr, cA, out, off, 16384,
        wsWioux, nullptr, wsWat, bioux, biouh, nullptr,
        wsBext, h_ext, Wa, 1, 0);

    // internal levels: 8192 .. 2 (13 levels), stream order = dependency
    int prevOff = off;
    off += 16384;
    float* cIn = cA; float* cOutP = cB;
    for (int n = 8192; n >= 2; n >>= 1) {
        int blocks = (n + 15) / 16;
        tree_level_kernel<<<blocks, 256, 0, stream>>>(
            nullptr, out + (size_t)prevOff * 128, cIn, cOutP, out, off, n,
            wsWiouh, wsWfh, wsWat, nullptr, biouh, bfh,
            wsBext, h_ext, Wa, 0, (n == 2) ? 1 : 0);
        prevOff = off;
        off += n;
        float* t = cIn; cIn = cOutP; cOutP = t;
    }
}